// MultiHeadAggregator_55825984913804
// MI455X (gfx1250) — compile-verified
//
#include <hip/hip_runtime.h>
#include <math.h>

typedef __attribute__((ext_vector_type(2))) float v2f;
typedef __attribute__((ext_vector_type(8))) float v8f;

#define LAYERS 8
#define BATCH  16
#define TLEN   2048
#define HDIM   768
#define CDIM   4
#define XPAD   772   // HDIM + 4 dwords: stride % 64 banks == 4 -> conflict-free column reads

// ---------------------------------------------------------------------------
// Kernel 1: per batch row, find first hte token position and build compact
// index lists for the desc mask and the hte window.
// ---------------------------------------------------------------------------
__global__ __launch_bounds__(256) void prep_kernel(
    const int* __restrict__ ids,       // (B,T)
    const int* __restrict__ hte_id_p,  // scalar
    const int* __restrict__ d_ids,     // (n_d)
    int n_d,
    const int* __restrict__ window_p,  // scalar
    int* __restrict__ desc_idx,        // (B,T)
    int* __restrict__ desc_cnt,        // (B)
    int* __restrict__ hte_idx,         // (B,8)
    int* __restrict__ hte_cnt)         // (B)
{
    __shared__ int s_pos, s_cnt;
    const int b = blockIdx.x, tid = threadIdx.x;
    const int hte_id = hte_id_p[0];
    const int window = window_p[0];
    if (tid == 0) { s_pos = TLEN; s_cnt = 0; }
    __syncthreads();
    const int* row = ids + (size_t)b * TLEN;
    for (int p = tid; p < TLEN; p += 256) {
        const int v = row[p];
        if (v == hte_id) atomicMin(&s_pos, p);
        bool m = false;
        for (int k = 0; k < n_d; ++k) m |= (v == d_ids[k]);
        if (m) {
            const int j = atomicAdd(&s_cnt, 1);
            desc_idx[(size_t)b * TLEN + j] = p;   // order irrelevant for softmax-pool
        }
    }
    __syncthreads();
    if (tid == 0) {
        int cnt = s_cnt;
        if (cnt == 0) { desc_idx[(size_t)b * TLEN] = 0; cnt = 1; }  // fallback: position 0
        desc_cnt[b] = cnt;
        const int pos = (s_pos == TLEN) ? (TLEN - 1) : s_pos;
        int nh = 0;
        for (int o = 0; o <= window; ++o) {
            const int wp = pos + o;
            if (wp < TLEN) hte_idx[b * 8 + nh++] = wp;   // invalid -> excluded == masked out
        }
        hte_cnt[b] = nh;   // >= 1 always (pos < T)
    }
}

// ---------------------------------------------------------------------------
// Kernel 2: masked attention pool over a compact position list.
// One block per (l,b); 8 wave32's. Scores via V_WMMA_F32_16X16X4_F32,
// online softmax over 16-row tiles, weighted sum in VALU.
// Writes pooled[l][b][c][h] (pre layer-weighting).
// ---------------------------------------------------------------------------
__global__ __launch_bounds__(256) void pool_kernel(
    const float* __restrict__ hs,      // (9,B,T,H): layer l maps to hs[l+1]
    const float* __restrict__ Wm,      // (4,768)
    const int* __restrict__ lists,     // (B, list_stride)
    const int* __restrict__ counts,    // (B)
    int list_stride,
    float* __restrict__ pooled_out)    // (L,B,C*H)
{
    __shared__ float xt[16 * XPAD];        // gathered tile rows
    __shared__ float sc[16][16];           // WMMA score tile (only cols 0..3 meaningful)
    __shared__ float pooled[CDIM][HDIM];   // online-softmax weighted accumulators
    __shared__ float wexp[16][CDIM];       // exp(score - runmax) per tile row
    __shared__ float run_max[CDIM], run_den[CDIM], scale_c[CDIM];
    __shared__ int   rowpos[16];

    const int tid = threadIdx.x;
    const int l   = blockIdx.x / BATCH;
    const int b   = blockIdx.x % BATCH;
    const int n   = counts[b];
    const int* list = lists + (size_t)b * list_stride;
    const float* xbase = hs + ((size_t)(l + 1) * BATCH + b) * (size_t)TLEN * HDIM;

    for (int i = tid; i < CDIM * HDIM; i += 256) pooled[i / HDIM][i % HDIM] = 0.0f;
    if (tid < CDIM) { run_max[tid] = -INFINITY; run_den[tid] = 0.0f; }
    __syncthreads();

    const int ntiles = (n + 15) >> 4;
    const int wave  = tid >> 5;
    const int lane  = tid & 31;
    const int mrow  = lane & 15;               // M row / N col for this lane
    const int khalf = (lane >> 4) << 1;        // 0 or 2: which K pair this lane holds
    const float* wrow = Wm + (size_t)(mrow & 3) * HDIM;   // clamped B-operand row
    const bool  bvalid = (mrow < CDIM);

    for (int t = 0; t < ntiles; ++t) {
        const int cnt = min(16, n - t * 16);
        if (tid < 16) rowpos[tid] = (tid < cnt) ? list[t * 16 + tid] : 0;
        sc[tid >> 4][tid & 15] = 0.0f;
        __syncthreads();

        // gather tile rows into LDS (float4, 16B aligned: 768*4 % 16 == 0)
        for (int e = tid; e < 16 * (HDIM / 4); e += 256) {
            const int r  = e / (HDIM / 4);
            const int h4 = e % (HDIM / 4);
            float4 v = make_float4(0.f, 0.f, 0.f, 0.f);
            if (r < cnt) {
                const float4* src = (const float4*)(xbase + (size_t)rowpos[r] * HDIM);
                v = src[h4];
            }
            *(float4*)&xt[r * XPAD + h4 * 4] = v;
        }
        __syncthreads();

        // scores[16 pos][16 cols] = X_tile (16xH) * W^T (Hx16, cols>=4 zero)
        // 192 K-chunks of 4 split across 8 waves; EXEC all-1s (uniform control flow)
        v8f acc = {};
        for (int kc = wave * 24; kc < wave * 24 + 24; ++kc) {
            const int kb = kc * 4 + khalf;
            v2f a, bb;
            a.x = xt[mrow * XPAD + kb];
            a.y = xt[mrow * XPAD + kb + 1];
            const float b0 = wrow[kb], b1 = wrow[kb + 1];
            bb.x = bvalid ? b0 : 0.0f;
            bb.y = bvalid ? b1 : 0.0f;
            acc = __builtin_amdgcn_wmma_f32_16x16x4_f32(
                false, a, false, bb, (short)0, acc, false, false);
        }
        const int moff = (lane >> 4) << 3;     // C/D layout: M = v + 8*(lane>=16)
        #pragma unroll
        for (int v = 0; v < 8; ++v) atomicAdd(&sc[v + moff][mrow], acc[v]);
        __syncthreads();

        // online softmax update per head (4 threads, <=16 rows each)
        if (tid < CDIM) {
            float tmax = -INFINITY;
            for (int j = 0; j < cnt; ++j) tmax = fmaxf(tmax, sc[j][tid]);
            const float nmax = fmaxf(run_max[tid], tmax);
            const float scl  = __expf(run_max[tid] - nmax);   // exp(-inf)=0 on first tile
            float den = run_den[tid] * scl;
            for (int j = 0; j < 16; ++j) {
                const float e = (j < cnt) ? __expf(sc[j][tid] - nmax) : 0.0f;
                wexp[j][tid] = e;
                den += e;
            }
            run_max[tid] = nmax;
            run_den[tid] = den;
            scale_c[tid] = scl;
        }
        __syncthreads();

        // pooled[c][h] = pooled[c][h]*scale + sum_j wexp[j][c]*xt[j][h]
        for (int h = tid; h < HDIM; h += 256) {
            #pragma unroll
            for (int c = 0; c < CDIM; ++c) {
                float s = 0.0f;
                #pragma unroll
                for (int j = 0; j < 16; ++j) s = fmaf(wexp[j][c], xt[j * XPAD + h], s);
                pooled[c][h] = pooled[c][h] * scale_c[c] + s;
            }
        }
        __syncthreads();
    }

    float* outp = pooled_out + ((size_t)l * BATCH + b) * (CDIM * HDIM);
    for (int i = tid; i < CDIM * HDIM; i += 256)
        outp[i] = pooled[i / HDIM][i % HDIM] / run_den[i / HDIM];
}

// ---------------------------------------------------------------------------
// Kernel 3: out[b, 0:3072]    = sum_l wl[l]*pooled_hte[l][b]
//           out[b, 3072:6144] = sum_l wl[l]*pooled_desc[l][b]
// ---------------------------------------------------------------------------
__global__ __launch_bounds__(256) void combine_kernel(
    const float* __restrict__ layer_logits,
    const float* __restrict__ pooled_hte,
    const float* __restrict__ pooled_desc,
    float* __restrict__ out)
{
    const int CH = CDIM * HDIM;   // 3072
    const int o = blockIdx.x * 256 + threadIdx.x;
    if (o >= BATCH * 2 * CH) return;
    float lg[LAYERS];
    float mx = -INFINITY;
    #pragma unroll
    for (int i = 0; i < LAYERS; ++i) { lg[i] = layer_logits[i]; mx = fmaxf(mx, lg[i]); }
    float den = 0.0f;
    #pragma unroll
    for (int i = 0; i < LAYERS; ++i) { lg[i] = __expf(lg[i] - mx); den += lg[i]; }
    const int b = o / (2 * CH);
    const int r = o % (2 * CH);
    const float* src = (r < CH) ? pooled_hte : pooled_desc;
    const int ch = (r < CH) ? r : r - CH;
    float acc = 0.0f;
    #pragma unroll
    for (int l = 0; l < LAYERS; ++l)
        acc += lg[l] * src[((size_t)l * BATCH + b) * CH + ch];
    out[o] = acc / den;
}

extern "C" void kernel_launch(void* const* d_in, const int* in_sizes, int n_in,
                              void* d_out, int out_size, void* d_ws, size_t ws_size,
                              hipStream_t stream) {
    const float* hs     = (const float*)d_in[0];   // (9,16,2048,768) f32
    const int*   ids    = (const int*)  d_in[1];   // (16,2048) i32
    const float* logits = (const float*)d_in[2];   // (8,) f32
    const float* Wm     = (const float*)d_in[3];   // (4,768) f32
    const int*   hteid  = (const int*)  d_in[4];   // scalar
    const int*   d_ids  = (const int*)  d_in[5];   // (3,)
    const int*   windp  = (const int*)  d_in[6];   // scalar
    const int n_d = in_sizes[5];

    char* ws = (char*)d_ws;
    int*   desc_idx    = (int*)ws;   ws += (size_t)BATCH * TLEN * sizeof(int);
    int*   desc_cnt    = (int*)ws;   ws += (size_t)BATCH * sizeof(int);
    int*   hte_idx     = (int*)ws;   ws += (size_t)BATCH * 8 * sizeof(int);
    int*   hte_cnt     = (int*)ws;   ws += (size_t)BATCH * sizeof(int);
    float* pooled_hte  = (float*)ws; ws += (size_t)LAYERS * BATCH * CDIM * HDIM * sizeof(float);
    float* pooled_desc = (float*)ws;

    prep_kernel<<<BATCH, 256, 0, stream>>>(ids, hteid, d_ids, n_d, windp,
                                           desc_idx, desc_cnt, hte_idx, hte_cnt);

    pool_kernel<<<LAYERS * BATCH, 256, 0, stream>>>(hs, Wm, hte_idx,  hte_cnt,  8,
                                                    pooled_hte);
    pool_kernel<<<LAYERS * BATCH, 256, 0, stream>>>(hs, Wm, desc_idx, desc_cnt, TLEN,
                                                    pooled_desc);

    const int total = BATCH * 2 * CDIM * HDIM;     // 98304 == out_size
    combine_kernel<<<(total + 255) / 256, 256, 0, stream>>>(logits, pooled_hte,
                                                            pooled_desc, (float*)d_out);
}